// MyEmbedding_39917426049447
// MI455X (gfx1250) — compile-verified
//
#include <hip/hip_runtime.h>

// Embedding lookup: out[t, :] = W[:, x[t]] + b
//   x: [819200] int32 indices, W: [128, 100000] f32, b: [128] f32
// Strategy (memory-bound, ~576 MB HBM traffic => ~25us at 23.3 TB/s):
//   1) build_table: Wt'[v][e] = W[e][v] + b[e]   (LDS tile transpose, bias folded in)
//   2) gather:      out[t]    = Wt'[x[t]]        (512B row copy per wave32;
//                    CDNA5 async Global<->LDS DMA when available, NT stores otherwise)

#define EMBED 128
#define ROW_BYTES (EMBED * 4)   // 512 B per embedding row
#define NBUF 4                  // rows in flight per wave
#define WAVES_PER_BLOCK 8       // 256 threads

typedef __attribute__((ext_vector_type(4))) float v4f;
typedef __attribute__((ext_vector_type(4))) int   v4i;

// Builtin signature (from clang diagnostic): first param is int4 in AS(1)
// ("__device__" is how HIP clang prints addrspace(1)); second is the LDS side.
typedef __attribute__((address_space(1))) v4i* gv4i_p;   // global int4*
typedef __attribute__((address_space(3))) v4i* lv4i_p;   // LDS int4*

#ifndef __has_builtin
#define __has_builtin(x) 0
#endif

#if __has_builtin(__builtin_amdgcn_global_load_async_to_lds_b128) && \
    __has_builtin(__builtin_amdgcn_global_store_async_from_lds_b128)
#define USE_ASYNC_LDS 1
#else
#define USE_ASYNC_LDS 0
#endif

__device__ __forceinline__ void wait_async0() {
#if __has_builtin(__builtin_amdgcn_s_wait_asynccnt)
    __builtin_amdgcn_s_wait_asynccnt(0);
#else
    asm volatile("s_wait_asynccnt 0" ::: "memory");
#endif
}

// ---------------------------------------------------------------------------
// Pass 1: Wt'[v*128 + e] = W[e*V + v] + b[e].  32x32 LDS tile transpose.
// Grid: ((V+31)/32, EMBED/32), block (32, 8).
// ---------------------------------------------------------------------------
__global__ void build_table(const float* __restrict__ W, const float* __restrict__ b,
                            float* __restrict__ Wt, int V) {
    __shared__ float tile[32][33];  // +1 pad: conflict-free transposed reads
    const int vBase = blockIdx.x * 32;
    const int eBase = blockIdx.y * 32;
    const int tx = threadIdx.x;     // 0..31
    const int ty = threadIdx.y;     // 0..7

    // Load: coalesced along v (contiguous in W's rows)
#pragma unroll
    for (int i = 0; i < 4; ++i) {
        const int e = eBase + ty + i * 8;
        const int v = vBase + tx;
        tile[ty + i * 8][tx] = (v < V) ? W[(size_t)e * V + v] : 0.0f;
    }
    __syncthreads();

    const float bias = b[eBase + tx];   // e = eBase + tx on the store side
    // Store: coalesced along e (contiguous in Wt's rows), bias folded in
#pragma unroll
    for (int i = 0; i < 4; ++i) {
        const int v = vBase + ty + i * 8;
        if (v < V)
            Wt[(size_t)v * EMBED + eBase + tx] = tile[tx][ty + i * 8] + bias;
    }
}

// ---------------------------------------------------------------------------
// Pass 2: out[t] = Wt'[x[t]].  One wave32 per token: 32 lanes x 16 B = 512 B.
// Async path: row -> LDS -> out via CDNA5 async DMA (ASYNCcnt), 4 rows in
// flight per wave, fenced with s_wait_asynccnt.  Fallback: B128 + NT B128.
// ---------------------------------------------------------------------------
__global__ void gather_rows(const int* __restrict__ x, const float* __restrict__ Wt,
                            float* __restrict__ out, int numTokens) {
#if USE_ASYNC_LDS
    __shared__ char stage[WAVES_PER_BLOCK][NBUF][ROW_BYTES];
#endif
    const int lane = threadIdx.x & 31;
    const int wave = threadIdx.x >> 5;
    const int waveId = blockIdx.x * WAVES_PER_BLOCK + wave;
    const int wavesTotal = gridDim.x * WAVES_PER_BLOCK;

    for (int base = waveId * NBUF; base < numTokens; base += wavesTotal * NBUF) {
#if USE_ASYNC_LDS
        // Stage NBUF embedding rows into LDS (ASYNCcnt++ each, no VGPR data)
#pragma unroll
        for (int i = 0; i < NBUF; ++i) {
            const int t = base + i;               // wave-uniform guard
            if (t < numTokens) {
                const int idx = x[t];
                char* src = (char*)Wt + ((size_t)idx * ROW_BYTES + lane * 16);
                char* dst = &stage[wave][i][lane * 16];
                __builtin_amdgcn_global_load_async_to_lds_b128(
                    (gv4i_p)src, (lv4i_p)dst, 0, 0);
            }
        }
        wait_async0();  // rows resident in LDS
        // Drain LDS -> output (stores also tracked by ASYNCcnt)
#pragma unroll
        for (int i = 0; i < NBUF; ++i) {
            const int t = base + i;
            if (t < numTokens) {
                char* dst = (char*)out + ((size_t)t * ROW_BYTES + lane * 16);
                char* src = &stage[wave][i][lane * 16];
                __builtin_amdgcn_global_store_async_from_lds_b128(
                    (gv4i_p)dst, (lv4i_p)src, 0, 0);
            }
        }
        wait_async0();  // LDS buffers safe to reuse next iteration
#else
#pragma unroll
        for (int i = 0; i < NBUF; ++i) {
            const int t = base + i;
            if (t < numTokens) {
                const int idx = x[t];
                v4f v = *(const v4f*)(Wt + (size_t)idx * EMBED + lane * 4);
                // NT store: don't let the 419MB output stream rinse the
                // L2-resident (51MB of 192MB) embedding table.
                __builtin_nontemporal_store(v, (v4f*)(out + (size_t)t * EMBED + lane * 4));
            }
        }
#endif
    }
}

// ---------------------------------------------------------------------------
// Fallback if workspace can't hold the transposed table: direct strided gather.
// ---------------------------------------------------------------------------
__global__ void gather_direct(const int* __restrict__ x, const float* __restrict__ W,
                              const float* __restrict__ b, float* __restrict__ out,
                              int numTokens, int V) {
    const size_t total = (size_t)numTokens * EMBED;
    for (size_t g = blockIdx.x * (size_t)blockDim.x + threadIdx.x; g < total;
         g += (size_t)gridDim.x * blockDim.x) {
        const int t = (int)(g >> 7);
        const int e = (int)(g & (EMBED - 1));
        out[g] = W[(size_t)e * V + x[t]] + b[e];
    }
}

extern "C" void kernel_launch(void* const* d_in, const int* in_sizes, int n_in,
                              void* d_out, int out_size, void* d_ws, size_t ws_size,
                              hipStream_t stream) {
    const int*   x = (const int*)d_in[0];
    const float* W = (const float*)d_in[1];
    const float* b = (const float*)d_in[2];
    float*       out = (float*)d_out;

    const int numTokens = in_sizes[0];
    const int V = in_sizes[1] / EMBED;  // 100000
    const size_t tableBytes = (size_t)V * EMBED * sizeof(float);  // 51.2 MB

    if (ws_size >= tableBytes) {
        float* Wt = (float*)d_ws;
        dim3 tb(32, 8);
        dim3 tg((V + 31) / 32, EMBED / 32);
        build_table<<<tg, tb, 0, stream>>>(W, b, Wt, V);

        const int tokensPerBlock = WAVES_PER_BLOCK * NBUF;  // 32
        int blocks = (numTokens + tokensPerBlock - 1) / tokensPerBlock;
        if (blocks > 8192) blocks = 8192;   // persistent, grid-stride
        gather_rows<<<blocks, WAVES_PER_BLOCK * 32, 0, stream>>>(x, Wt, out, numTokens);
    } else {
        gather_direct<<<4096, 256, 0, stream>>>(x, W, b, out, numTokens, V);
    }
}